// SpermLossEMA_43138651521211
// MI455X (gfx1250) — compile-verified
//
#include <hip/hip_runtime.h>
#include <cstdint>

#define NB 4096
#define NH 512
#define NT 16
#define HT 8192  // NH*NT floats per b-row (= 32 KB)

__device__ __forceinline__ float chan_of(int t) {
  if (t == 13) return 0.0f;
  if (t == 11 || t == 12) return 1.0f;
  return 0.1f;
}

// One workgroup per b. Stage pred[b]/targ[b] (64 KB) into LDS with CDNA5
// async-to-LDS copies (non-temporal: the 268 MB stream must not thrash the
// 192 MB L2, where the reused 32 KB weights table lives), then compute all
// three partial sums from LDS in a single DRAM pass.
__global__ __launch_bounds__(256) void sperm_main_kernel(
    const float* __restrict__ pred, const float* __restrict__ targ,
    const float* __restrict__ wts, float* __restrict__ partial) {
  __shared__ __align__(16) float lp[HT];
  __shared__ __align__(16) float lt[HT];
  __shared__ float rbuf[8];

  const int tid = threadIdx.x;
  const int b = blockIdx.x;
  const float* gp = pred + (size_t)b * HT + tid * 4;
  const float* gt = targ + (size_t)b * HT + tid * 4;

  // ---- Async stage: one base address per array, per-chunk stride via the
  // 24-bit INST_OFFSET (ISA applies it to BOTH the global and LDS address).
  const uint32_t ldsp = (uint32_t)(uintptr_t)(const void*)&lp[tid * 4];
  const uint32_t ldst = (uint32_t)(uintptr_t)(const void*)&lt[tid * 4];
#pragma unroll
  for (int i = 0; i < 8; ++i) {
    asm volatile("global_load_async_to_lds_b128 %0, %1, off offset:%2 th:TH_LOAD_NT"
                 :: "v"(ldsp), "v"(gp), "i"(i * 4096) : "memory");
    asm volatile("global_load_async_to_lds_b128 %0, %1, off offset:%2 th:TH_LOAD_NT"
                 :: "v"(ldst), "v"(gt), "i"(i * 4096) : "memory");
  }

  // Overlap the (L2-resident) weights fetch with the async LDS staging.
  float4 wv[8];
#pragma unroll
  for (int i = 0; i < 8; ++i) wv[i] = *(const float4*)&wts[(i * 256 + tid) * 4];

  asm volatile("s_wait_asynccnt 0x0" ::: "memory");
  __syncthreads();

  // ---- Weighted loss + a0 partials. Each thread owns the same 4 t-lanes in
  // every chunk, so the chan factors are loop-invariant.
  const int t0 = (tid & 3) * 4;
  const float c0 = chan_of(t0 + 0), c1 = chan_of(t0 + 1);
  const float c2 = chan_of(t0 + 2), c3 = chan_of(t0 + 3);
  float wsum = 0.f, a0 = 0.f;
#pragma unroll
  for (int i = 0; i < 8; ++i) {
    const int e = (i * 256 + tid) * 4;  // == h*16 + t, also the weights index
    const float4 p = *(const float4*)&lp[e];
    const float4 q = *(const float4*)&lt[e];
    const float4 w = wv[i];
    const float d0 = p.x - q.x, d1 = p.y - q.y;
    const float d2 = p.z - q.z, d3 = p.w - q.w;
    const float l0 = d0 * d0 * c0, l1 = d1 * d1 * c1;
    const float l2 = d2 * d2 * c2, l3 = d3 * d3 * c3;
    wsum += l0 * w.x + l1 * w.y + l2 * w.z + l3 * w.w;
    if (i == 0 && tid == 0) {  // e==0 -> h=0, t=0..2 (chan=0.1 already in l*)
      a0 += (l0 + 1e-10f) / (w.x + 1e-10f);
      a0 += (l1 + 1e-10f) / (w.y + 1e-10f);
      a0 += (l2 + 1e-10f) / (w.z + 1e-10f);
    }
  }

  // ---- EMA along H (pred ch13 vs targ ch3), wave0 only: affine-map scan.
  // e[h] = 0.25*x[h] + 0.75*e[h-1], e[0] = x[0] (seed carry with x0).
  float esum = 0.f;
  if (tid < 32) {
    const float r = 0.75f, al = 0.25f;
    float M = 1.f, C = 0.f;  // lane's segment as map e -> M*e + C
#pragma unroll
    for (int j = 0; j < 16; ++j) {
      const float x = lp[(tid * 16 + j) * NT + 13];
      C = al * x + r * C;
      M *= r;
    }
    for (int d = 1; d < 32; d <<= 1) {  // inclusive scan of map composition
      const float Mp = __shfl_up(M, d, 32);
      const float Cp = __shfl_up(C, d, 32);
      if (tid >= d) { C = M * Cp + C; M = M * Mp; }
    }
    const float x0 = lp[13];
    const float Min = __shfl_up(M, 1, 32);
    const float Cin = __shfl_up(C, 1, 32);
    float e = (tid == 0) ? x0 : (Min * x0 + Cin);  // carry into this segment
#pragma unroll
    for (int j = 0; j < 16; ++j) {
      const int h = tid * 16 + j;
      const float x = lp[h * NT + 13];
      e = al * x + r * e;
      const float dd = e - lt[h * NT + 3];
      esum += dd * dd;
    }
    for (int d = 16; d > 0; d >>= 1) esum += __shfl_xor(esum, d, 32);
  }

  // ---- Deterministic block reduce of wsum; write per-block partials.
  for (int d = 16; d > 0; d >>= 1) wsum += __shfl_xor(wsum, d, 32);
  if ((tid & 31) == 0) rbuf[tid >> 5] = wsum;
  __syncthreads();
  if (tid == 0) {
    float tot = 0.f;
#pragma unroll
    for (int wv8 = 0; wv8 < 8; ++wv8) tot += rbuf[wv8];
    partial[b * 3 + 0] = tot;
    partial[b * 3 + 1] = esum;  // tid0 holds the full wave0 esum
    partial[b * 3 + 2] = a0;
  }
}

// Fixed-order final reduction over the 4096 per-block partials.
__global__ __launch_bounds__(256) void sperm_final_kernel(
    const float* __restrict__ partial, float* __restrict__ out) {
  __shared__ float rb[24];
  const int tid = threadIdx.x;
  float s0 = 0.f, s1 = 0.f, s2 = 0.f;
  for (int i = tid; i < NB; i += 256) {
    s0 += partial[i * 3 + 0];
    s1 += partial[i * 3 + 1];
    s2 += partial[i * 3 + 2];
  }
  for (int d = 16; d > 0; d >>= 1) {
    s0 += __shfl_xor(s0, d, 32);
    s1 += __shfl_xor(s1, d, 32);
    s2 += __shfl_xor(s2, d, 32);
  }
  if ((tid & 31) == 0) {
    const int wv = tid >> 5;
    rb[wv * 3 + 0] = s0; rb[wv * 3 + 1] = s1; rb[wv * 3 + 2] = s2;
  }
  __syncthreads();
  if (tid == 0) {
    float t0 = 0.f, t1 = 0.f, t2 = 0.f;
#pragma unroll
    for (int wv = 0; wv < 8; ++wv) {
      t0 += rb[wv * 3 + 0]; t1 += rb[wv * 3 + 1]; t2 += rb[wv * 3 + 2];
    }
    out[0] = t0 / (float)((double)NB * NH * NT) + t1 / (float)((double)NB * NH);
    out[1] = t2 / (float)(NB * 3);
  }
}

extern "C" void kernel_launch(void* const* d_in, const int* in_sizes, int n_in,
                              void* d_out, int out_size, void* d_ws, size_t ws_size,
                              hipStream_t stream) {
  const float* pred = (const float*)d_in[0];   // (4096, 512, 16) f32
  const float* targ = (const float*)d_in[1];   // (4096, 512, 16) f32
  const float* wts  = (const float*)d_in[2];   // (512, 16) f32
  // d_in[3] = action_dim (==3), compile-time constant here.
  float* partial = (float*)d_ws;               // 4096*3 floats = 48 KB
  float* out = (float*)d_out;                  // {total, a0_loss}

  sperm_main_kernel<<<NB, 256, 0, stream>>>(pred, targ, wts, partial);
  sperm_final_kernel<<<1, 256, 0, stream>>>(partial, out);
}